// CSWinBlock_8985071583249
// MI455X (gfx1250) — compile-verified
//
#include <hip/hip_runtime.h>
#include <hip/hip_bf16.h>

// Fused CSWin block for MI455X (gfx1250, wave32).
// Memory-bound: 268 MB min traffic @ 23.3 TB/s ~= 11.5 us; ~6 GFLOP total.
// fp32 end-to-end; 16-dim matmuls via V_WMMA_F32_16X16X4_F32 (4 chained K=4 steps).
// __launch_bounds__(128, 4): ~5 waves/SIMD for HBM latency hiding (need ~16MB
// in flight; ~2KB NT loads per wave). Attention phase split per-branch with a
// barrier fence + LN folded into the store loop to shrink peak live registers
// so the allocator stays spill-free under the cap.

typedef __attribute__((ext_vector_type(2))) float v2f;
typedef __attribute__((ext_vector_type(4))) float v4f;
typedef __attribute__((ext_vector_type(8))) float v8f;

#define C16       16
#define ROWS      8
#define COLS      16
#define TOKENS    128          // tokens per block (8 rows x 16 cols of token grid)
#define NTHREADS  128          // 4 waves of 32
#define WGRID     512          // token grid is 512 x 512 (Hg = Wg = 2048/4)
#define NTOK      (512 * 512)  // tokens per batch
#define TS        20           // LDS row stride (16-ch rows), 16B aligned, conflict-free
#define QS        52           // LDS row stride (48-ch qkv rows)

__global__ __launch_bounds__(NTHREADS, 4)
void cswin_fused_kernel(const float* __restrict__ x,
                        const float* __restrict__ qkv_w,   // (16,48) row-major: q|k|v col blocks
                        const float* __restrict__ cw0, const float* __restrict__ cb0,
                        const float* __restrict__ cw1, const float* __restrict__ cb1,
                        const float* __restrict__ proj_w,  // (16,16)
                        const float* __restrict__ proj_b,
                        const float* __restrict__ ln_w, const float* __restrict__ ln_b,
                        float* __restrict__ out)
{
    __shared__ float s_tok[TOKENS * TS];   // input tokens (residual source + WMMA A)
    __shared__ float s_qkv[TOKENS * QS];   // q(0..15) k(16..31) v(32..47)
    __shared__ float s_att[TOKENS * TS];   // branch concat (proj input / proj output)

    const int tid = threadIdx.x;
    const int b   = blockIdx.z;
    const int gy0 = blockIdx.y * ROWS;     // token-grid row of tile origin (multiple of 8)
    const int gx0 = blockIdx.x * COLS;     // token-grid col of tile origin (multiple of 16)

    const float* __restrict__ xb = x + (size_t)b * NTOK * C16;

    // ---- Stage 128 tokens x 16 ch into LDS (16B vectors, coalesced, non-temporal) ----
    #pragma unroll
    for (int it = 0; it < 4; ++it) {
        const int i   = tid + it * NTHREADS;       // 512 16-byte transfers
        const int row = i >> 6;                    // 64 transfers per token-grid row
        const int j   = i & 63;
        const v4f val = __builtin_nontemporal_load(reinterpret_cast<const v4f*>(
            xb + ((size_t)(gy0 + row) * WGRID + gx0) * C16 + j * 4));
        const int tk = row * COLS + (j >> 2);
        *reinterpret_cast<v4f*>(&s_tok[tk * TS + (j & 3) * 4]) = val;
    }
    __syncthreads();

    const int lane = tid & 31;
    const int wave = tid >> 5;
    const int m    = lane & 15;   // row (A/D) or column (B/D) index within 16
    const int half = lane >> 4;   // lane half per ISA 7.12.2 layouts

    // ---- qkv = tok @ qkv_w via WMMA f32 16x16x4 (K=16 -> 4 steps) ----
    // A frag: a[j] = A[m][ks + 2*half + j]; B frag: b[j] = B[ks + 2*half + j][m]
    // D frag: d[r] -> D[rowbase + r + 8*half][m]
    #pragma unroll 1
    for (int ci = 0; ci < 2; ++ci) {
        const int rowbase = (wave + ci * 4) * 16;
        #pragma unroll
        for (int s = 0; s < 3; ++s) {              // q, k, v output tiles
            v8f acc = {0.f, 0.f, 0.f, 0.f, 0.f, 0.f, 0.f, 0.f};
            #pragma unroll
            for (int ks = 0; ks < 16; ks += 4) {
                const int kk = ks + 2 * half;
                v2f a, bb;
                a.x  = s_tok[(rowbase + m) * TS + kk];
                a.y  = s_tok[(rowbase + m) * TS + kk + 1];
                bb.x = qkv_w[kk * 48 + s * 16 + m];
                bb.y = qkv_w[(kk + 1) * 48 + s * 16 + m];
                acc = __builtin_amdgcn_wmma_f32_16x16x4_f32(
                        false, a, false, bb, (short)0, acc, false, false);
            }
            #pragma unroll
            for (int r = 0; r < 8; ++r)
                s_qkv[(rowbase + r + 8 * half) * QS + s * 16 + m] = acc[r];
        }
    }
    __syncthreads();

    // ---- Window attention + LePE, one query token per thread ----
    const float scale = 0.35355339059327373f;   // 8^-0.5

    // Branch 1: horizontal windows of 8 consecutive tokens, channels 0..7
    {
        const int wbase = tid & ~7;              // window start token (row-local)
        const int lx    = tid & 7;               // position inside window
        float q[8];
        #pragma unroll
        for (int c = 0; c < 8; ++c) q[c] = s_qkv[tid * QS + c];
        float sc[8]; float mx = -1e30f;
        #pragma unroll
        for (int j = 0; j < 8; ++j) {
            const float* kr = &s_qkv[(wbase + j) * QS + 16];
            float s = 0.f;
            #pragma unroll
            for (int c = 0; c < 8; ++c) s += q[c] * kr[c];
            s *= scale; sc[j] = s; mx = fmaxf(mx, s);
        }
        float den = 0.f;
        #pragma unroll
        for (int j = 0; j < 8; ++j) { sc[j] = __expf(sc[j] - mx); den += sc[j]; }
        const float inv = 1.0f / den;
        #pragma unroll
        for (int c = 0; c < 8; ++c) {
            float acc = 0.f;
            #pragma unroll
            for (int j = 0; j < 8; ++j) acc += sc[j] * s_qkv[(wbase + j) * QS + 32 + c];
            // depthwise 3x3 on a 1x8 window image -> 3-tap horizontal conv (middle kernel row)
            float lep = cb0[c] + cw0[c * 9 + 4] * s_qkv[tid * QS + 32 + c];
            if (lx > 0) lep += cw0[c * 9 + 3] * s_qkv[(tid - 1) * QS + 32 + c];
            if (lx < 7) lep += cw0[c * 9 + 5] * s_qkv[(tid + 1) * QS + 32 + c];
            s_att[tid * TS + c] = acc * inv + lep;   // store immediately, free regs
        }
    }
    __syncthreads();   // scheduling fence: keeps branch-2 LDS prefetch out of branch 1

    // Branch 2: vertical windows of 8 rows (same column), channels 8..15
    {
        const int col = tid & 15;
        const int ly  = tid >> 4;                // row inside tile == row inside window
        float q[8];
        #pragma unroll
        for (int c = 0; c < 8; ++c) q[c] = s_qkv[tid * QS + 8 + c];
        float sc[8]; float mx = -1e30f;
        #pragma unroll
        for (int j = 0; j < 8; ++j) {
            const float* kr = &s_qkv[(j * COLS + col) * QS + 24];
            float s = 0.f;
            #pragma unroll
            for (int c = 0; c < 8; ++c) s += q[c] * kr[c];
            s *= scale; sc[j] = s; mx = fmaxf(mx, s);
        }
        float den = 0.f;
        #pragma unroll
        for (int j = 0; j < 8; ++j) { sc[j] = __expf(sc[j] - mx); den += sc[j]; }
        const float inv = 1.0f / den;
        #pragma unroll
        for (int c = 0; c < 8; ++c) {
            float acc = 0.f;
            #pragma unroll
            for (int j = 0; j < 8; ++j) acc += sc[j] * s_qkv[(j * COLS + col) * QS + 40 + c];
            // depthwise 3x3 on an 8x1 window image -> 3-tap vertical conv (middle kernel col)
            float lep = cb1[c] + cw1[c * 9 + 4] * s_qkv[tid * QS + 40 + c];
            if (ly > 0) lep += cw1[c * 9 + 1] * s_qkv[(tid - COLS) * QS + 40 + c];
            if (ly < 7) lep += cw1[c * 9 + 7] * s_qkv[(tid + COLS) * QS + 40 + c];
            s_att[tid * TS + 8 + c] = acc * inv + lep;
        }
    }
    __syncthreads();

    // ---- proj: att @ proj_w + proj_b via WMMA (accumulator seeded with bias) ----
    #pragma unroll 1
    for (int ci = 0; ci < 2; ++ci) {
        const int rowbase = (wave + ci * 4) * 16;
        const float bias  = proj_b[m];
        v8f acc;
        #pragma unroll
        for (int r = 0; r < 8; ++r) acc[r] = bias;
        #pragma unroll
        for (int ks = 0; ks < 16; ks += 4) {
            const int kk = ks + 2 * half;
            v2f a, bb;
            a.x  = s_att[(rowbase + m) * TS + kk];
            a.y  = s_att[(rowbase + m) * TS + kk + 1];
            bb.x = proj_w[kk * 16 + m];
            bb.y = proj_w[(kk + 1) * 16 + m];
            acc = __builtin_amdgcn_wmma_f32_16x16x4_f32(
                    false, a, false, bb, (short)0, acc, false, false);
        }
        #pragma unroll
        for (int r = 0; r < 8; ++r)
            s_att[(rowbase + r + 8 * half) * TS + m] = acc[r];   // rows owned by this wave
    }
    __syncthreads();

    // ---- LayerNorm(16) + residual, fused into the transposed store loop ----
    float vr[16]; float mu = 0.f;
    #pragma unroll
    for (int c = 0; c < 16; ++c) { vr[c] = s_att[tid * TS + c]; mu += vr[c]; }
    mu *= (1.0f / 16.0f);
    float var = 0.f;
    #pragma unroll
    for (int c = 0; c < 16; ++c) { const float d = vr[c] - mu; var += d * d; }
    var *= (1.0f / 16.0f);
    const float rstd = rsqrtf(var + 1e-5f);

    // out[b][c*N + n], n = token index; consecutive lanes -> consecutive n (coalesced)
    const size_t n     = (size_t)(gy0 + (tid >> 4)) * WGRID + gx0 + (tid & 15);
    float* __restrict__ ob = out + (size_t)b * NTOK * C16;
    #pragma unroll
    for (int c = 0; c < 16; ++c) {
        const float r = s_tok[tid * TS + c] + (vr[c] - mu) * rstd * ln_w[c] + ln_b[c];
        __builtin_nontemporal_store(r, &ob[(size_t)c * NTOK + n]);
    }
}

extern "C" void kernel_launch(void* const* d_in, const int* in_sizes, int n_in,
                              void* d_out, int out_size, void* d_ws, size_t ws_size,
                              hipStream_t stream) {
    (void)in_sizes; (void)n_in; (void)out_size; (void)d_ws; (void)ws_size;
    const float* x      = (const float*)d_in[0];
    const float* qkv_w  = (const float*)d_in[1];
    const float* cw0    = (const float*)d_in[2];
    const float* cb0    = (const float*)d_in[3];
    const float* cw1    = (const float*)d_in[4];
    const float* cb1    = (const float*)d_in[5];
    const float* proj_w = (const float*)d_in[6];
    const float* proj_b = (const float*)d_in[7];
    const float* ln_w   = (const float*)d_in[8];
    const float* ln_b   = (const float*)d_in[9];
    float* out = (float*)d_out;

    dim3 grid(WGRID / COLS, WGRID / ROWS, 8);   // 32 x 64 x 8 = 16384 blocks
    dim3 block(NTHREADS);
    hipLaunchKernelGGL(cswin_fused_kernel, grid, block, 0, stream,
                       x, qkv_w, cw0, cb0, cw1, cb1, proj_w, proj_b, ln_w, ln_b, out);
}